// RelGCN_59365037965371
// MI455X (gfx1250) — compile-verified
//
#include <hip/hip_runtime.h>

#define N_NODES 50000
#define N_EDGES 500000
#define D 128
#define SQRT_ALPHA 0.70710678118654752f   // sqrt(0.5)
#define SQRT_1MA   0.70710678118654752f   // sqrt(1-0.5)

typedef __attribute__((ext_vector_type(2))) float v2f;
typedef __attribute__((ext_vector_type(8))) float v8f;

// ---------------------------------------------------------------------------
// Kernel 1: per-relation node transform + loop message, via fp32 WMMA.
//   h_rel[t][n][:] = sqrt(alpha)   * feat[n][:] @ weight[t]    (t = 0,1) -> ws
//   out[n][:]      = sqrt(1-alpha) * feat[n][:] @ loop_weight + bias     -> d_out
// Block = 256 threads (8 wave32), owns 16 rows of feat staged in LDS.
// 24 jobs = {W0,W1,Wloop} x 8 column tiles of 16; 3 jobs per wave.
// ---------------------------------------------------------------------------
__global__ void __launch_bounds__(256)
rgcn_gemm_kernel(const float* __restrict__ feat,
                 const float* __restrict__ weight,      // [2,128,128]
                 const float* __restrict__ loop_weight, // [128,128]
                 const float* __restrict__ h_bias,      // [128]
                 float* __restrict__ hrel,              // ws: [2,N,128]
                 float* __restrict__ out)               // [N,128]
{
    __shared__ float a_lds[16 * 130];                  // padded stride: no bank conflicts

    const int m0  = blockIdx.x << 4;                   // 16-row tile (50000 % 16 == 0)
    const int tid = threadIdx.x;

    // Stage A tile: feat[m0 .. m0+15][0..127] -> LDS (coalesced)
    #pragma unroll
    for (int idx = tid; idx < 16 * 128; idx += 256) {
        const int r = idx >> 7;
        const int c = idx & 127;
        a_lds[r * 130 + c] = feat[(size_t)(m0 + r) * D + c];
    }
    __syncthreads();

    const int wave = tid >> 5;
    const int lane = tid & 31;
    const int l15  = lane & 15;
    const int hi   = lane >> 4;                        // 0: K%4 in {0,1}, 1: K%4 in {2,3}

    // A fragment source for this lane (ISA 16x4 f32 A layout)
    const float* abase = &a_lds[l15 * 130 + 2 * hi];

    for (int j = wave; j < 24; j += 8) {               // wave-uniform job loop
        const int t  = j >> 3;                         // 0,1 = relations; 2 = loop
        const int n0 = (j & 7) << 4;                   // output column tile
        const float* w = (t < 2) ? (weight + (size_t)t * D * D) : loop_weight;

        const int col = n0 + l15;
        const float* bbase = w + (size_t)(2 * hi) * D + col;   // B frag: rows k, k+1

        v8f acc = {0.f, 0.f, 0.f, 0.f, 0.f, 0.f, 0.f, 0.f};

        #pragma unroll
        for (int kk = 0; kk < 32; ++kk) {
            v2f a, b;
            a.x = abase[4 * kk];
            a.y = abase[4 * kk + 1];
            b.x = bbase[(size_t)(4 * kk) * D];
            b.y = bbase[(size_t)(4 * kk + 1) * D];
            // D = A(16x4,f32) * B(4x16,f32) + C ; full fp32 accumulate
            acc = __builtin_amdgcn_wmma_f32_16x16x4_f32(
                /*neg_a=*/false, a, /*neg_b=*/false, b,
                /*c_mod=*/(short)0, acc, /*reuse_a=*/false, /*reuse_b=*/false);
        }

        // C/D layout: acc[v] -> row m0 + v + 8*hi, col n0 + (lane&15)
        const int mb = m0 + (hi << 3);
        if (t < 2) {
            float* p = hrel + ((size_t)t * N_NODES + mb) * D + col;
            #pragma unroll
            for (int v = 0; v < 8; ++v)
                p[(size_t)v * D] = acc[v] * SQRT_ALPHA;     // fold sqrt(alpha) here
        } else {
            const float bb = h_bias[col];
            float* p = out + (size_t)mb * D + col;
            #pragma unroll
            for (int v = 0; v < 8; ++v)
                p[(size_t)v * D] = acc[v] * SQRT_1MA + bb;  // loop msg + bias
        }
    }
}

// ---------------------------------------------------------------------------
// Kernel 2: edge gather + scatter-sum.
// One wave32 per edge; lane handles 4 contiguous floats (float4 load from
// L2-resident hrel, 4 non-returning global f32 atomics into d_out[dst]).
// hrel already carries the sqrt(alpha) scale.
// ---------------------------------------------------------------------------
__global__ void __launch_bounds__(256)
rgcn_edge_kernel(const float* __restrict__ hrel,
                 const int* __restrict__ src,
                 const int* __restrict__ dst,
                 const int* __restrict__ etypes,
                 float* __restrict__ out)
{
    const int e = blockIdx.x * 8 + (threadIdx.x >> 5);
    if (e >= N_EDGES) return;
    const int lane = threadIdx.x & 31;

    const int s = src[e];
    const int d = dst[e];
    const int t = etypes[e];

    const float4 v = *reinterpret_cast<const float4*>(
        hrel + ((size_t)t * N_NODES + (size_t)s) * D + lane * 4);

    float* o = out + (size_t)d * D + lane * 4;
    atomicAdd(o + 0, v.x);
    atomicAdd(o + 1, v.y);
    atomicAdd(o + 2, v.z);
    atomicAdd(o + 3, v.w);
}

// ---------------------------------------------------------------------------
extern "C" void kernel_launch(void* const* d_in, const int* in_sizes, int n_in,
                              void* d_out, int out_size, void* d_ws, size_t ws_size,
                              hipStream_t stream)
{
    const float* feat        = (const float*)d_in[0];
    const int*   src         = (const int*)  d_in[1];
    const int*   dst         = (const int*)  d_in[2];
    const int*   etypes      = (const int*)  d_in[3];
    const float* weight      = (const float*)d_in[4];
    const float* loop_weight = (const float*)d_in[5];
    const float* h_bias      = (const float*)d_in[6];
    float*       out         = (float*)d_out;
    float*       hrel        = (float*)d_ws;   // needs 2*50000*128*4 = 51.2 MB

    // Phase 1: WMMA transforms (also fully initializes d_out with loop+bias)
    rgcn_gemm_kernel<<<N_NODES / 16, 256, 0, stream>>>(
        feat, weight, loop_weight, h_bias, hrel, out);

    // Phase 2: edge gather + atomic scatter-sum (one wave per edge)
    rgcn_edge_kernel<<<(N_EDGES + 7) / 8, 256, 0, stream>>>(
        hrel, src, dst, etypes, out);
}